// DifferentiableExtrusion_43301860278300
// MI455X (gfx1250) — compile-verified
//
#include <hip/hip_runtime.h>
#include <hip/hip_bf16.h>
#include <math.h>

typedef __attribute__((ext_vector_type(2))) float v2f;
typedef __attribute__((ext_vector_type(8))) float v8f;

#define VOX   96
#define GRIDP (VOX*VOX)     // 9216 grid points
#define TILES (GRIDP/16)    // 576 point-tiles per batch
#define NB    4
#define NPOLY 32
#define NVERT 32
#define SHARP 100.0f
#define EPSF  1e-8f

// XOR-butterfly step on 8 lanes-worth of values via ds_swizzle_b32
// (group-of-32 mode: offset[14:10]=xor_mask, offset[4:0]=and_mask).
// Single LDS-hw instruction per value, no index/address VALU setup.
template <int M>
__device__ __forceinline__ void min_swizzle_step(float (&v)[8]) {
#pragma unroll
  for (int r = 0; r < 8; ++r) {
    const int o = __builtin_amdgcn_ds_swizzle(__float_as_int(v[r]),
                                              (M << 10) | 0x1f);
    v[r] = fminf(v[r], __int_as_float(o));
  }
}

// ---------------------------------------------------------------------------
// Phase 1: soft-SDF polygon masks.  One wave32 = one (batch, 16-point tile,
// polygon-group).  NGRP is a compile-time constant so the wave decode is
// mul-shift and the polygon loop has a constant trip count.  96 = 6*16, so a
// tile never crosses a grid row: py is wave-uniform.  Per polygon & 16-edge
// tile, two v_wmma_f32_16x16x4_f32 (K=2 padded to 4, full fp32):
//   D1 = (p - v0).e       (A=(px,py), B=e,      C=-v0.e)
//   D2 = |p - v0|^2       (A=(px,py), B=-2*v0,  C=|p|^2+|v0|^2)
// Crossing parity via compare-ballots + scalar popcounts (bits 0-15 = point
// r, bits 16-31 = point 8+r); min-distance uses a ds_swizzle XOR butterfly.
// ---------------------------------------------------------------------------
template <int NGRP>
__global__ __launch_bounds__(256) void masks_kernel(
    const float* __restrict__ poly,        // (B,N,P,2)
    const float* __restrict__ validity,    // (B,N)
    float*       __restrict__ comb_out)    // (NGRP,B,V*V) scratch
{
  constexpr int NPER = NPOLY / NGRP;
  const int lane = threadIdx.x & 31;
  const int wid  = threadIdx.x >> 5;
  const int gw   = blockIdx.x * 8 + wid;         // NB*TILES*NGRP waves
  const int b    = gw / (TILES * NGRP);          // constant divisor
  const int rem  = gw % (TILES * NGRP);
  const int tIdx = rem % TILES;
  const int grp  = rem / TILES;
  const int nBeg = grp * NPER;

  const int p0      = tIdx * 16;                 // first grid point of tile
  const int col     = lane & 15;                 // edge column owned by lane
  const int rowBase = (lane < 16) ? 0 : 8;       // D-matrix row half
  const float inv   = 1.0f / (float)(VOX - 1);
  const float loMul = (lane < 16) ? 1.0f : 0.0f; // zero-pad K=2,3 half

  // Wave-uniform y for the whole tile; per-lane x values.
  const int   gx0 = p0 % VOX;                    // tile never crosses a row
  const float pyU = (float)(p0 / VOX) * inv;     // uniform -> SGPR math

  // A operand (16x4 f32): lanes 0-15 hold row m=lane, K=0/1; lanes 16-31 = 0.
  v2f amat;
  amat.x = loMul * ((float)(gx0 + col) * inv);
  amat.y = loMul * pyU;

  // Points this lane sees in the D matrix: rows rowBase..rowBase+7, col=col.
  float px[8], pp[8];
#pragma unroll
  for (int r = 0; r < 8; ++r) {
    px[r] = (float)(gx0 + rowBase + r) * inv;
    pp[r] = px[r] * px[r] + pyU * pyU;
  }

  float comb[8];
#pragma unroll
  for (int r = 0; r < 8; ++r) comb[r] = 0.0f;

  const float* polyB = poly + ((size_t)b * NPOLY) * NVERT * 2;

  for (int i = 0; i < NPER; ++i) {
    const int n = nBeg + i;
    if (i + 1 < NPER)                             // cover next polygon fetch
      __builtin_prefetch(polyB + (size_t)(n + 1) * NVERT * 2 + 4 * col, 0, 3);

    // Scalar validity test: s_cmp + s_cbranch, EXEC stays all-1s for WMMA.
    const float vs = __int_as_float(
        __builtin_amdgcn_readfirstlane(__float_as_int(validity[b * NPOLY + n])));
    if (vs < 0.5f) continue;

    float minD2[8];
    int   cLo[8], cHi[8];                         // crossing counts (uniform)
#pragma unroll
    for (int r = 0; r < 8; ++r) { minD2[r] = 3.0e38f; cLo[r] = 0; cHi[r] = 0; }

#pragma unroll
    for (int et = 0; et < 2; ++et) {             // 32 edges = 2 tiles of 16
      const int k  = et * 16 + col;
      const int k1 = (k + 1) & (NVERT - 1);      // roll(-1) wraps mod P
      const float* vp = polyB + (size_t)n * NVERT * 2;
      const float x0 = vp[2 * k + 0],  y0 = vp[2 * k + 1];
      const float x1 = vp[2 * k1 + 0], y1 = vp[2 * k1 + 1];
      const float ex = x1 - x0, ey = y1 - y0;
      const float ee   = ex * ex + ey * ey;               // |e|^2
      const float resq = __builtin_amdgcn_rcpf(ee + EPSF);
      const float v0e  = x0 * ex + y0 * ey;               // v0 . e
      const float v00  = x0 * x0 + y0 * y0;               // |v0|^2

      // Crossing helpers: py is uniform, so these are per-edge only.
      const float rdy = __builtin_amdgcn_rcpf(ey + EPSF); // 1/(y1-y0+EPS)
      const float ix  = fmaf(ex, (pyU - y0) * rdy, x0);   // inter_x
      // Straddling-edge mask: XOR of two bare compare-ballots (no i1
      // materialization; two v_cmp into SGPRs + one s_xor).
      const unsigned ycm = __builtin_amdgcn_ballot_w32(y0 <= pyU) ^
                           __builtin_amdgcn_ballot_w32(y1 <= pyU);

      // B operands (4x16 f32): K=0/1 rows in lanes 0-15, zeros in 16-31.
      v2f bmatE, bmatV;
      bmatE.x = loMul * ex;
      bmatE.y = loMul * ey;
      bmatV.x = loMul * (-2.0f * x0);
      bmatV.y = loMul * (-2.0f * y0);

      // C operands: per-edge constant / per-row bias folded into the MACs.
      v8f cE, cV;
#pragma unroll
      for (int r = 0; r < 8; ++r) { cE[r] = -v0e; cV[r] = pp[r] + v00; }

      v8f dT = __builtin_amdgcn_wmma_f32_16x16x4_f32(  // (p-v0).e
          false, amat, false, bmatE, (short)0, cE, false, false);
      v8f dQ = __builtin_amdgcn_wmma_f32_16x16x4_f32(  // |p-v0|^2
          false, amat, false, bmatV, (short)0, cV, false, false);

#pragma unroll
      for (int r = 0; r < 8; ++r) {
        const float tnum = dT[r];
        const float t    = fminf(fmaxf(tnum * resq, 0.0f), 1.0f);
        // d2 = |p-v0|^2 - 2 t tnum + t^2 |e|^2
        const float d2 = fmaf(t, fmaf(t, ee, -(tnum + tnum)), dQ[r]);
        minD2[r] = fminf(minD2[r], d2);

        // Bare compare-ballot folds to one v_cmp into SGPR; AND with the
        // per-tile edge mask.  bits 0-15 -> point r, bits 16-31 -> point 8+r.
        const unsigned m = __builtin_amdgcn_ballot_w32(ix > px[r]) & ycm;
        cLo[r] += __builtin_popcount(m & 0xFFFFu);   // scalar s_bcnt + s_add
        cHi[r] += __builtin_popcount(m >> 16);
      }
    }

    // Min-reduce over the 16 edge columns via ds_swizzle XOR butterfly
    // (xor 1,2,4,8 stays inside each lane half).
    min_swizzle_step<1>(minD2);
    min_swizzle_step<2>(minD2);
    min_swizzle_step<4>(minD2);
    min_swizzle_step<8>(minD2);

#pragma unroll
    for (int r = 0; r < 8; ++r) {
      const float dmin = __builtin_amdgcn_sqrtf(fmaxf(minD2[r], 0.0f));
      const int   cnt  = (lane < 16) ? cLo[r] : cHi[r];
      const float sdf  = (cnt & 1) ? -dmin : dmin;
      const float msk  = 1.0f / (1.0f + __expf(sdf * SHARP));
      comb[r] = fmaxf(comb[r], msk);
    }
  }

  // After the butterfly all 16 lanes of a half agree; lane col==0 of each
  // half writes its 8 points into this group's slice.
  if (col == 0) {
#pragma unroll
    for (int r = 0; r < 8; ++r)
      comb_out[((size_t)grp * NB + b) * GRIDP + p0 + rowBase + r] = comb[r];
  }
}

// ---------------------------------------------------------------------------
// Phase 2: depth extrusion — streaming broadcast with the cross-group max
// folded in (fully unrolled).  Write-bound: 14.2 MB stores vs <=0.6 MB reads
// @ 23.3 TB/s.   out[b][z][y][x] = max_g comb[g][b][y][x] * (z < hv_b)
// ---------------------------------------------------------------------------
template <int NGRP>
__global__ __launch_bounds__(256) void extrude_kernel(
    const float* __restrict__ comb,       // (NGRP, B, V*V)
    const float* __restrict__ attrs,      // (B, 8)
    float*       __restrict__ out)        // (B, V, V, V)
{
  const long long VOL  = (long long)VOX * VOX * VOX;
  const long long tid  = (long long)blockIdx.x * blockDim.x + threadIdx.x;
  const long long base = tid * 4;
  if (base >= (long long)NB * VOL) return;

  const int b   = (int)(base / VOL);
  const int rem = (int)(base % VOL);
  const int z   = rem / GRIDP;
  const int yx  = rem % GRIDP;            // 96 % 4 == 0 -> stays in one row

  const float a0 = attrs[b * 8];
  const float hv = fminf(fmaxf(rintf(fminf(fmaxf(a0, 0.0f), 1.0f) * (float)VOX),
                               1.0f), (float)VOX);
  const float d  = ((float)z < hv) ? 1.0f : 0.0f;

  float4 c = *(const float4*)(comb + (size_t)b * GRIDP + yx);
#pragma unroll
  for (int g = 1; g < NGRP; ++g) {
    const float4 cg = *(const float4*)(comb + ((size_t)g * NB + b) * GRIDP + yx);
    c.x = fmaxf(c.x, cg.x); c.y = fmaxf(c.y, cg.y);
    c.z = fmaxf(c.z, cg.z); c.w = fmaxf(c.w, cg.w);
  }

  float4 o;
  o.x = c.x * d; o.y = c.y * d; o.z = c.z * d; o.w = c.w * d;
  *(float4*)(out + base) = o;
}

// ---------------------------------------------------------------------------
extern "C" void kernel_launch(void* const* d_in, const int* in_sizes, int n_in,
                              void* d_out, int out_size, void* d_ws, size_t ws_size,
                              hipStream_t stream) {
  const float* poly     = (const float*)d_in[0];   // (4,32,32,2) f32
  const float* attrs    = (const float*)d_in[1];   // (4,8)       f32
  const float* validity = (const float*)d_in[2];   // (4,32)      f32
  float* out  = (float*)d_out;                     // (4,96,96,96) f32
  float* comb = (float*)d_ws;                      // (NGRP,4,9216) f32 scratch

  // 4-way polygon split (4x waves = 9216, 1/4 sequential depth) when the
  // scratch fits.  Deterministic: depends only on the harness-fixed ws_size.
  if (ws_size >= (size_t)4 * NB * GRIDP * sizeof(float)) {
    masks_kernel<4><<<(NB * TILES * 4) / 8, 256, 0, stream>>>(poly, validity, comb);
    extrude_kernel<4><<<3456, 256, 0, stream>>>(comb, attrs, out);
  } else {
    masks_kernel<1><<<(NB * TILES * 1) / 8, 256, 0, stream>>>(poly, validity, comb);
    extrude_kernel<1><<<3456, 256, 0, stream>>>(comb, attrs, out);
  }
}